// RoIAlign_28991029248096
// MI455X (gfx1250) — compile-verified
//
#include <hip/hip_runtime.h>

// RoIAlign for MI455X (gfx1250): memory-bound gather, TDM-pipelined.
//
// One 256-thread block per (roi, oy). For each 64-channel chunk, the two
// needed feature rows (y0,y1) over a 32-px x-window are staged into LDS by
// the Tensor Data Mover (tensor_load_to_lds, TENSORcnt): a 32x64 2D tile
// with channel stride H*W, hardware-padded 1 DWORD per 32 (stride-33 LDS
// layout -> bank-conflict-free compute reads). Double-buffered so chunk k+1
// streams in while chunk k is computed. Falls back to per-lane
// global_load_async_to_lds_b32 (ASYNCcnt) if the TDM builtin is absent.
// Output uses non-temporal stores (write-once stream; preserves L2
// residency of the 124.6MB feature tensor in the 192MB L2).

#define OH 7
#define OW 7
#define FC 256
#define FH 100
#define FW 152
#define CHUNK 64                  // channels per pipeline stage
#define NCH (FC / CHUNK)          // 4 stages
#define XT 32                     // staged x-window (ROI x-span <= 29 px)
#define XPAD 33                   // stride-33 layout (TDM pad: 1dw per 32dw)
#define BUFSZ (2 * CHUNK * XPAD)  // floats per buffer (4224 = 16,896 B)

#if defined(__has_builtin)
#if __has_builtin(__builtin_amdgcn_tensor_load_to_lds)
#define TDM_OK 1
#endif
#endif

__device__ __forceinline__ int imin(int a, int b) { return a < b ? a : b; }
__device__ __forceinline__ int imax(int a, int b) { return a > b ? a : b; }

#ifdef TDM_OK
typedef unsigned int v4u __attribute__((ext_vector_type(4)));
typedef int          v8i __attribute__((ext_vector_type(8)));
typedef int          v4i __attribute__((ext_vector_type(4)));

// Issue one TDM transfer: 2D tile (tile_dim0=32 x-elems, tile_dim1=64
// channels, stride tensor_dim0_stride=FH*FW) from global row start into LDS,
// with HW padding of 1 DWORD after every 32 DWORDs -> stride-33 LDS lines.
__device__ __forceinline__ void tdm_stage_row(const float* gsrc,
                                              unsigned lds_byte_addr,
                                              int width /* FW - xloC */)
{
    const unsigned long long ga = (unsigned long long)(uintptr_t)gsrc;
    v4u g0;
    g0.x = 1u;                                            // count=1 (user D#)
    g0.y = lds_byte_addr;                                 // LDS dest (bytes)
    g0.z = (unsigned)(ga & 0xFFFFFFFFu);                  // global addr lo
    g0.w = (unsigned)((ga >> 32) & 0x1FFFFFFu)            // global addr hi25
         | (2u << 30);                                    // type=2 ("image")
    v8i g1;
    g1[0] = (2 << 16)      // data_size = 4 bytes
          | (1 << 20)      // pad_enable
          | (4 << 22);     // pad_interval code 4 = every 32 DWORDs
                           // pad_amount field 0 = 1 DWORD; wg_mask=0
    g1[1] = (width & 0xFFFF) << 16;        // tensor_dim0 lo16 (abar addr = 0)
    g1[2] = ((width >> 16) & 0xFFFF)       // tensor_dim0 hi16
          | (CHUNK << 16);                 // tensor_dim1 lo16 = 64
    g1[3] = (XT << 16);                    // tensor_dim1 hi16=0; tile_dim0=32
    g1[4] = CHUNK;                         // tile_dim1=64; tile_dim2=0
    g1[5] = FH * FW;                       // tensor_dim0_stride lo32 = 15200
    g1[6] = 0;                             // stride hi16; dim1_stride lo16
    g1[7] = 0;                             // dim1_stride hi32 (unused, 2D)
    const v4i z4 = {0, 0, 0, 0};           // groups 2/3 unused (<=2D tile)
    const v8i z8 = {0, 0, 0, 0, 0, 0, 0, 0};
    // 6-arg form (clang-23 / therock-10.0 headers):
    // (g0:v4u, g1:v8i, g2:v4i, g3:v4i, v8i, cpol:i32)
    __builtin_amdgcn_tensor_load_to_lds(g0, g1, z4, z4, z8, 0);
}
#endif

__global__ __launch_bounds__(256) void roi_align_kernel(
    const float* __restrict__ feat, const float* __restrict__ rois,
    float* __restrict__ out, int R)
{
    __shared__ float tile[2 * BUFSZ];   // 33,792 B: double buffer

    const int bid = blockIdx.x;
    const int r  = bid / OH;
    const int oy = bid % OH;
    if (r >= R) return;

    // ROI params (uniform per block -> scalarized)
    const float* roi = rois + (size_t)r * 5;
    const int   bi  = (int)roi[0];
    const float rx1 = roi[1], ry1 = roi[2], rx2 = roi[3], ry2 = roi[4];
    const float bw = (rx2 - rx1) / (float)OW;
    const float bh = (ry2 - ry1) / (float)OH;

    // y interpolation state for this oy (uniform)
    const float gy  = ry1 + ((float)oy + 0.5f) * bh;
    const float iy  = ((gy + 1.0f) * (float)FH - 1.0f) * 0.5f;
    const float y0f = floorf(iy);
    const float wy1 = iy - y0f;
    const float wy0 = 1.0f - wy1;
    const int y0i = (int)y0f;
    const int y1i = y0i + 1;
    const float vy0 = (y0i >= 0 && y0i < FH) ? 1.0f : 0.0f;
    const float vy1 = (y1i >= 0 && y1i < FH) ? 1.0f : 0.0f;
    const int yA = imin(imax(y0i, 0), FH - 1);
    const int yB = imin(imax(y1i, 0), FH - 1);

    // x window start (uniform): first corner of bin 0
    const float ix0 = ((rx1 + 0.5f * bw + 1.0f) * (float)FW - 1.0f) * 0.5f;
    const int xlo  = (int)floorf(ix0);
    const int xloC = imax(xlo, 0);

    const int wave = threadIdx.x >> 5;
    const int lane = threadIdx.x & 31;
    const int xg   = imin(xloC + lane, FW - 1);   // (fallback path)

    // LDS byte address base (generic LDS pointer low 32 bits = LDS offset)
    const unsigned lds_base = (unsigned)(uintptr_t)&tile[0];

    auto issue_stage = [&](int ch, int bsel) {
        const int cbase = ch * CHUNK;
#ifdef TDM_OK
        if (wave == 0) {
            const float* pA =
                feat + (((size_t)bi * FC + (size_t)cbase) * FH + yA) * (size_t)FW + xloC;
            const float* pB =
                feat + (((size_t)bi * FC + (size_t)cbase) * FH + yB) * (size_t)FW + xloC;
            const int width = FW - xloC;
            tdm_stage_row(pA, lds_base + 4u * (unsigned)(bsel * BUFSZ), width);
            tdm_stage_row(pB, lds_base + 4u * (unsigned)(bsel * BUFSZ + CHUNK * XPAD), width);
        }
#else
        // Fallback: per-lane CDNA5 async gather, 2*CHUNK coalesced 128B rows.
        for (int task = wave; task < 2 * CHUNK; task += 8) {
            const int cl  = task & (CHUNK - 1);
            const int row = task >> 6;               // CHUNK == 64
            const int y   = row ? yB : yA;
            const float* gsrc =
                feat + (((size_t)bi * FC + (size_t)(cbase + cl)) * FH + y) * (size_t)FW
                     + (size_t)xg;
            const unsigned ldsaddr =
                lds_base + 4u * (unsigned)(bsel * BUFSZ + (row * CHUNK + cl) * XPAD + lane);
            asm volatile("global_load_async_to_lds_b32 %0, %1, off"
                         :: "v"(ldsaddr), "v"(gsrc) : "memory");
        }
#endif
    };

    issue_stage(0, 0);

    for (int ch = 0; ch < NCH; ++ch) {
        // My stage-ch writes have landed in LDS:
#ifdef TDM_OK
        if (wave == 0) {
#if __has_builtin(__builtin_amdgcn_s_wait_tensorcnt)
            __builtin_amdgcn_s_wait_tensorcnt(0);
#else
            asm volatile("s_wait_tensorcnt 0" ::: "memory");
#endif
        }
#else
        asm volatile("s_wait_asynccnt 0" ::: "memory");
#endif
        // Everyone's stage-ch landed AND everyone finished compute(ch-1),
        // so the other buffer is free to overwrite:
        __syncthreads();

        // Kick off next chunk's staging; it flies during compute below.
        if (ch + 1 < NCH) issue_stage(ch + 1, (ch + 1) & 1);

        const float* buf   = &tile[(ch & 1) * BUFSZ];
        const int    cbase = ch * CHUNK;

        // compute: 7 bins x CHUNK channels from LDS
        for (int e = threadIdx.x; e < OW * CHUNK; e += 256) {
            const int ox = e >> 6;             // uniform within a wave
            const int cl = e & (CHUNK - 1);    // consecutive across lanes

            const float gx  = rx1 + ((float)ox + 0.5f) * bw;
            const float ix  = ((gx + 1.0f) * (float)FW - 1.0f) * 0.5f;
            const float x0f = floorf(ix);
            const float wxb = ix - x0f;
            const float wxa = 1.0f - wxb;
            const int x0i = (int)x0f;
            const int x1i = x0i + 1;
            const float vx0 = (x0i >= 0 && x0i < FW) ? 1.0f : 0.0f;
            const float vx1 = (x1i >= 0 && x1i < FW) ? 1.0f : 0.0f;

            const int sx0 = imin(imax(imin(imax(x0i, 0), FW - 1) - xloC, 0), XT - 1);
            const int sx1 = imin(imax(imin(imax(x1i, 0), FW - 1) - xloC, 0), XT - 1);

            const float* t0 = &buf[cl * XPAD];
            const float* t1 = &buf[(CHUNK + cl) * XPAD];
            const float v00 = t0[sx0], v01 = t0[sx1];
            const float v10 = t1[sx0], v11 = t1[sx1];

            const float w00 = wy0 * wxa * vy0 * vx0;
            const float w01 = wy0 * wxb * vy0 * vx1;
            const float w10 = wy1 * wxa * vy1 * vx0;
            const float w11 = wy1 * wxb * vy1 * vx1;

            const float res = ((v00 * w00 + v01 * w01) + v10 * w10) + v11 * w11;

            const size_t oidx =
                (((size_t)r * OH + oy) * OW + ox) * (size_t)FC + (size_t)(cbase + cl);
            __builtin_nontemporal_store(res, out + oidx);
        }
        // No trailing barrier needed: next iteration's wait+barrier orders
        // buffer reuse (stage ch+2 is only issued after that barrier).
    }
}

extern "C" void kernel_launch(void* const* d_in, const int* in_sizes, int n_in,
                              void* d_out, int out_size, void* d_ws, size_t ws_size,
                              hipStream_t stream)
{
    const float* features = (const float*)d_in[0];
    const float* rois     = (const float*)d_in[1];
    float* out            = (float*)d_out;

    const int R = in_sizes[1] / 5;          // 2048
    dim3 grid((unsigned)(R * OH));          // one block per (roi, oy)
    dim3 block(256);
    roi_align_kernel<<<grid, block, 0, stream>>>(features, rois, out, R);
}